// TUS_80891414053505
// MI455X (gfx1250) — compile-verified
//
#include <hip/hip_runtime.h>
#include <cstdint>

#define ALPHA_C 0.1f
#define C_  3
#define H_  540
#define W_  960
#define HO  1080
#define WO  1920

// Tile geometry: 36x128 output pixels -> 18x64 low-res + 1-halo = 20x66 LDS tile
#define TLH 20
#define TLW 66

__global__ __launch_bounds__(256) void taa_upsample_kernel(
    const float* __restrict__ frame,   // (4,3,540,960)
    const float* __restrict__ jitter,  // (4,2)
    float* __restrict__ out)           // (4,3,1080,1920)
{
    __shared__ float tile[TLH * TLW];

    const int tid   = threadIdx.x;
    const int tileX = blockIdx.x;   // 0..14
    const int tileY = blockIdx.y;   // 0..29
    const int nc    = blockIdx.z;   // 0..11
    const int n     = nc / 3;

    const int Y0 = tileY * 36;
    const int X0 = tileX * 128;
    const int rBase = (Y0 >> 1) - 1;   // may be -1 (edge-replicated)
    const int cBase = (X0 >> 1) - 1;

    const float* plane = frame + (size_t)nc * (H_ * W_);

    // ---- Stage low-res tile into LDS with CDNA5 async global->LDS loads ----
    // GVS addressing: 64-bit scalar base (SGPR pair) + 32-bit per-lane byte offset.
    const unsigned lds0 = (unsigned)(uintptr_t)(&tile[0]); // flat addr[31:0] == LDS byte offset
    const unsigned long long sbase = (unsigned long long)(uintptr_t)plane;
    for (int idx = tid; idx < TLH * TLW; idx += 256) {
        int r = idx / TLW;
        int c = idx - r * TLW;
        int gr = rBase + r; gr = gr < 0 ? 0 : (gr > H_ - 1 ? H_ - 1 : gr);
        int gc = cBase + c; gc = gc < 0 ? 0 : (gc > W_ - 1 ? W_ - 1 : gc);
        unsigned voff = ((unsigned)(gr * W_ + gc)) << 2;   // byte offset within plane
        unsigned dst  = lds0 + (unsigned)idx * 4u;
        asm volatile("global_load_async_to_lds_b32 %0, %1, %2"
                     :: "v"(dst), "v"(voff), "s"(sbase) : "memory");
    }
    asm volatile("s_wait_asynccnt 0" ::: "memory");
    __syncthreads();

    // ---- Per-thread invariants ----
    const float jx  = jitter[2 * n + 0];
    const float jy  = jitter[2 * n + 1];
    const int jinx  = (int)floorf(jx * 2.0f);          // z_up x parity (batch n)
    const int jiny  = (int)floorf(jy * 2.0f);          // z_up y parity (batch n)
    const int j0x   = (int)floorf(jitter[0] * 2.0f);   // beta parity (batch 0!)
    const int j0y   = (int)floorf(jitter[1] * 2.0f);

    const int lane_x  = tid & 127;
    const int row_off = tid >> 7;            // 0 or 1: fixed y parity per thread
    const int x  = X0 + lane_x;
    const int tc = (lane_x >> 1) + 1;        // tile col of (x>>1): 1..64

    // bilinear x: ix = (x>>1) + (x&1 ? .75 : .25) - jx  (constant per thread)
    float u  = ((x & 1) ? 0.75f : 0.25f) - jx;
    float fu = floorf(u);                    // -1 or 0
    const float wx    = u - fu;
    const bool  dxneg = (fu < 0.0f);
    // bilinear y: same trick, y parity fixed = row_off
    float v  = (row_off ? 0.75f : 0.25f) - jy;
    float fv = floorf(v);
    const float wy    = v - fv;
    const bool  dyneg = (fv < 0.0f);

    const bool zflag = (row_off == jiny) && ((x & 1) == jinx);
    const bool bflag = (row_off == j0y) && ((x & 1) == j0x);
    const float bwt  = bflag ? ALPHA_C : 0.0f;
    const float omb  = 1.0f - bwt;

    // per-row stats: 3 LDS loads -> row max/min, bilinear x-lerp, center value
    auto rowstat = [&](int r, float& rmax, float& rmin, float& lerp, float& cen) {
        const float c0 = tile[r * TLW + tc - 1];
        const float c1 = tile[r * TLW + tc];
        const float c2 = tile[r * TLW + tc + 1];
        rmax = fmaxf(fmaxf(c0, c1), c2);
        rmin = fminf(fminf(c0, c1), c2);
        cen  = c1;
        const float lo = dxneg ? c0 : c1;
        const float hi = dxneg ? c1 : c2;
        lerp = lo + wx * (hi - lo);
    };

    float amax, amin, alerp;
    float bmax, bmin, blerp, bcen;
    float dead;
    rowstat(0, amax, amin, alerp, dead);
    rowstat(1, bmax, bmin, blerp, bcen);

    float* po = out + (size_t)nc * (HO * WO) + (size_t)(Y0 + row_off) * WO + x;

    #pragma unroll
    for (int k = 0; k < 18; ++k) {
        float cmax, cmin, clerp, ccen;
        rowstat(k + 2, cmax, cmin, clerp, ccen);

        const float mx  = fmaxf(fmaxf(amax, bmax), cmax);   // 3x3 max pool
        const float mn  = fminf(fminf(amin, bmin), cmin);   // 3x3 min pool
        const float top = dyneg ? alerp : blerp;            // bilinear y-lerp
        const float bot = dyneg ? blerp : clerp;
        const float bil = top + wy * (bot - top);

        float hist = fmaxf(mn, fminf(mx, bil));             // clamp to neighborhood
        const float zz = zflag ? bcen : 0.0f;               // jitter-aligned zero-upsample
        __builtin_nontemporal_store(bwt * zz + omb * hist, po);  // beta blend, NT store
        po += 2 * WO;

        amax = bmax; amin = bmin; alerp = blerp;
        bmax = cmax; bmin = cmin; blerp = clerp; bcen = ccen;
    }
}

extern "C" void kernel_launch(void* const* d_in, const int* in_sizes, int n_in,
                              void* d_out, int out_size, void* d_ws, size_t ws_size,
                              hipStream_t stream) {
    const float* frame  = (const float*)d_in[0];   // (4,3,540,960) fp32
    const float* jitter = (const float*)d_in[1];   // (4,2) fp32
    float* out = (float*)d_out;                    // (4,3,1080,1920) fp32
    (void)in_sizes; (void)n_in; (void)out_size; (void)d_ws; (void)ws_size;

    dim3 grid(15, 30, 12);   // x-tiles, y-tiles, N*C planes
    taa_upsample_kernel<<<grid, 256, 0, stream>>>(frame, jitter, out);
}